// SparseMoE_19000935318137
// MI455X (gfx1250) — compile-verified
//
#include <hip/hip_runtime.h>

typedef __attribute__((ext_vector_type(2))) float v2f;
typedef __attribute__((ext_vector_type(8))) float v8f;

#define TILE_M 128
#define TILE_N 64
#define KB     32
#define KSTR   36   // padded LDS row stride (floats): 144B rows -> 16B aligned, bank-conflict free

// --------------------------------------------------------------------------
// LDS helpers: AS(3) pointer value == byte offset the DS/async unit adds to
// LDS_BASE, so cast generic -> AS3 and truncate.
// --------------------------------------------------------------------------
typedef __attribute__((address_space(3))) void lds_void;

__device__ __forceinline__ uint32_t lds_offset(void* p) {
    return (uint32_t)(uintptr_t)(lds_void*)p;
}

// CDNA5 async global->LDS copy (VGLOBAL encoding, tracked with ASYNCcnt).
// %0 = LDS byte-offset VGPR (VDST), %1 = 64-bit global address VGPR pair.
__device__ __forceinline__ void async_ld_b128(uint32_t ldsOff, const float* g) {
    asm volatile("global_load_async_to_lds_b128 %0, %1, off"
                 :: "v"(ldsOff), "v"(g)
                 : "memory");
}

__device__ __forceinline__ void wait_async0() {
#if __has_builtin(__builtin_amdgcn_s_wait_asynccnt)
    __builtin_amdgcn_s_wait_asynccnt(0);
#else
    asm volatile("s_wait_asynccnt 0" ::: "memory");
#endif
}

// ---------------------------------------------------------------------------
// Kernel 0: zero output buffer + per-expert counters
// ---------------------------------------------------------------------------
__global__ __launch_bounds__(256) void moe_zero(float* __restrict__ out, size_t n,
                                                int* __restrict__ counts, int E) {
    size_t i = (size_t)blockIdx.x * blockDim.x + threadIdx.x;
    size_t stride = (size_t)gridDim.x * blockDim.x;
    for (; i < n; i += stride) out[i] = 0.0f;
    if (blockIdx.x == 0 && (int)threadIdx.x < E) counts[threadIdx.x] = 0;
}

// ---------------------------------------------------------------------------
// Kernel 1: gating. One wave per token; softmax is monotonic so top-k on raw
// logits (+bias) gives identical routing. Appends token to its 2 experts'
// compacted index lists.
// ---------------------------------------------------------------------------
__global__ __launch_bounds__(256) void moe_gate(const float* __restrict__ x,
                                                const float* __restrict__ gw,
                                                const float* __restrict__ gb,
                                                int* __restrict__ counts,
                                                int* __restrict__ idxList,
                                                int T, int D) {
    const int wave = threadIdx.x >> 5;
    const int lane = threadIdx.x & 31;
    const int t = blockIdx.x * 8 + wave;
    if (t >= T) return;

    const float* xr = x + (size_t)t * D;
    float s[8];
#pragma unroll
    for (int e = 0; e < 8; ++e) s[e] = 0.0f;

    for (int k = lane; k < D; k += 32) {
        const float xv = xr[k];
#pragma unroll
        for (int e = 0; e < 8; ++e)
            s[e] = fmaf(xv, gw[e * D + k], s[e]);
    }
#pragma unroll
    for (int off = 16; off > 0; off >>= 1) {
#pragma unroll
        for (int e = 0; e < 8; ++e)
            s[e] += __shfl_xor(s[e], off, 32);
    }

    if (lane == 0) {
#pragma unroll
        for (int e = 0; e < 8; ++e) s[e] += gb[e];
        int i1 = 0; float v1 = s[0];
#pragma unroll
        for (int e = 1; e < 8; ++e) { if (s[e] > v1) { v1 = s[e]; i1 = e; } }
        int i2 = (i1 == 0) ? 1 : 0; float v2 = s[i2];
#pragma unroll
        for (int e = 0; e < 8; ++e) { if (e != i1 && s[e] > v2) { v2 = s[e]; i2 = e; } }
        int p1 = atomicAdd(&counts[i1], 1);
        idxList[i1 * T + p1] = t;
        int p2 = atomicAdd(&counts[i2], 1);
        idxList[i2 * T + p2] = t;
    }
}

// ---------------------------------------------------------------------------
// Kernel 2: grouped GEMM per expert, V_WMMA_F32_16X16X4_F32, async-to-LDS
// double-buffered staging. Block = 256 thr (8 waves) -> 128(M) x 64(N) tile.
// ---------------------------------------------------------------------------
__global__ __launch_bounds__(256) void moe_gemm(const float* __restrict__ x,      // [T,D]
                                                const float* __restrict__ W,      // [E,H,D]
                                                const float* __restrict__ bias,   // [E,H]
                                                const int* __restrict__ counts,   // [E]
                                                const int* __restrict__ idxList,  // [E,T]
                                                float* __restrict__ out,          // [T,H]
                                                int T, int D, int H) {
    const int e   = blockIdx.z;
    const int m0  = blockIdx.y * TILE_M;
    const int n0  = blockIdx.x * TILE_N;
    const int cnt = counts[e];
    if (m0 >= cnt) return;   // uniform scalar branch: whole block exits

    __shared__ alignas(16) float sX[2][TILE_M * KSTR];
    __shared__ alignas(16) float sW[2][TILE_N * KSTR];
    __shared__ int sTok[TILE_M];

    const int tid  = threadIdx.x;
    const int lane = tid & 31;
    const int wave = tid >> 5;
    const int l    = lane & 15;
    const int half = lane >> 4;

    if (tid < TILE_M) {
        const int r = m0 + tid;
        sTok[tid] = (r < cnt) ? idxList[e * T + r] : idxList[e * T];  // pad with safe row
    }
    __syncthreads();

    const float* We = W + (size_t)e * H * D;

    // Per-thread staging slots: 4 X rows + 2 W rows, fixed row/col4 per thread.
    const int rowBase = tid >> 3;   // 0..31   (8 float4 per 32-float row)
    const int c4      = tid & 7;

    const float* gx[4];
    uint32_t     lx[2][4];
#pragma unroll
    for (int i = 0; i < 4; ++i) {
        const int row = rowBase + 32 * i;            // 0..127
        gx[i]    = x + (size_t)sTok[row] * D + c4 * 4;
        lx[0][i] = lds_offset(&sX[0][row * KSTR + c4 * 4]);
        lx[1][i] = lds_offset(&sX[1][row * KSTR + c4 * 4]);
    }
    const float* gww[2];
    uint32_t     lw[2][2];
#pragma unroll
    for (int i = 0; i < 2; ++i) {
        const int row = rowBase + 32 * i;            // 0..63
        gww[i]   = We + (size_t)(n0 + row) * D + c4 * 4;
        lw[0][i] = lds_offset(&sW[0][row * KSTR + c4 * 4]);
        lw[1][i] = lds_offset(&sW[1][row * KSTR + c4 * 4]);
    }

    // Prologue: stage k-block 0 into buffer 0 (6 async b128 per thread).
#pragma unroll
    for (int i = 0; i < 4; ++i) async_ld_b128(lx[0][i], gx[i]);
#pragma unroll
    for (int i = 0; i < 2; ++i) async_ld_b128(lw[0][i], gww[i]);

    v8f acc[4] = {v8f{}, v8f{}, v8f{}, v8f{}};

    int buf = 0;
    for (int kb = 0; kb < D; kb += KB, buf ^= 1) {
        wait_async0();       // this wave's async writes have landed in LDS
        __syncthreads();     // everyone's have

        if (kb + KB < D) {   // stage next k-block into the other buffer
            const int nb = buf ^ 1;
#pragma unroll
            for (int i = 0; i < 4; ++i) { gx[i]  += KB; async_ld_b128(lx[nb][i], gx[i]); }
#pragma unroll
            for (int i = 0; i < 2; ++i) { gww[i] += KB; async_ld_b128(lw[nb][i], gww[i]); }
        }

        const float* Xc = sX[buf];
        const float* Wc = sW[buf];
#pragma unroll
        for (int kk = 0; kk < KB; kk += 4) {
            const int ko = kk + 2 * half;            // A: lanes<16 -> K=0,1; lanes>=16 -> K=2,3
            v2f a;
            a.x = Xc[(wave * 16 + l) * KSTR + ko];
            a.y = Xc[(wave * 16 + l) * KSTR + ko + 1];
#pragma unroll
            for (int nt = 0; nt < 4; ++nt) {
                v2f b;                               // B[k][n] = W[e][n][k]
                b.x = Wc[(nt * 16 + l) * KSTR + ko];
                b.y = Wc[(nt * 16 + l) * KSTR + ko + 1];
                acc[nt] = __builtin_amdgcn_wmma_f32_16x16x4_f32(
                    false, a, false, b, (short)0, acc[nt], false, false);
            }
        }
    }

    // Scatter: C layout M = v + 8*half, N = lane&15 per 16x16 tile. Two expert
    // blocks hit the same token row -> atomic f32 adds.
#pragma unroll
    for (int nt = 0; nt < 4; ++nt) {
        const int col  = n0 + nt * 16 + l;
        const float bc = bias[(size_t)e * H + col];
#pragma unroll
        for (int v = 0; v < 8; ++v) {
            const int rowInTile = wave * 16 + v + 8 * half;
            const int r = m0 + rowInTile;
            if (r < cnt) {
                const int tok = sTok[rowInTile];
                atomicAdd(&out[(size_t)tok * H + col], 0.5f * (acc[nt][v] + bc));
            }
        }
    }
}

// ---------------------------------------------------------------------------
extern "C" void kernel_launch(void* const* d_in, const int* in_sizes, int n_in,
                              void* d_out, int out_size, void* d_ws, size_t ws_size,
                              hipStream_t stream) {
    const float* x    = (const float*)d_in[0];   // sequences [N,P,D] == [T,D]
    const float* gw   = (const float*)d_in[1];   // [E,D]
    const float* gb   = (const float*)d_in[2];   // [E]
    const float* W    = (const float*)d_in[3];   // [E,H,D]
    const float* bias = (const float*)d_in[4];   // [E,H]
    float* out = (float*)d_out;

    const int E = in_sizes[2];
    const int D = in_sizes[1] / E;
    const int H = in_sizes[4] / E;
    const int T = in_sizes[0] / D;

    int* counts  = (int*)d_ws;          // [E] (padded to 64 ints)
    int* idxList = counts + 64;         // [E, T]

    moe_zero<<<4096, 256, 0, stream>>>(out, (size_t)out_size, counts, E);
    moe_gate<<<(T + 7) / 8, 256, 0, stream>>>(x, gw, gb, counts, idxList, T, D);

    dim3 grid(H / TILE_N, (T + TILE_M - 1) / TILE_M, E);
    moe_gemm<<<grid, 256, 0, stream>>>(x, W, bias, counts, idxList, out, T, D, H);
}